// MMD_36163624633049
// MI455X (gfx1250) — compile-verified
//
#include <hip/hip_runtime.h>
#include <hip/hip_bf16.h>

typedef __attribute__((ext_vector_type(2))) float v2f;
typedef __attribute__((ext_vector_type(8))) float v8f;

#define NTOT   4096   // 2*N
#define NHALF  2048
#define DDIM   64
#define TILE_M 128
#define TILE_N 64
#define LDS_STRIDE 68 // 64 + 4 pad: keeps 16B alignment, spreads LDS banks

// ---------------------------------------------------------------------------
// Kernel 1: per-batch row norms sq[i] = |T_i|^2, plus bandwidth via the
// identity  sum(L2) = 2*N*sum(sq) - 2*sum_d (sum_i T_id)^2   (O(N*D), no GEMM)
// ---------------------------------------------------------------------------
__global__ __launch_bounds__(256)
void mmd_prep(const float* __restrict__ x, const float* __restrict__ y,
              float* __restrict__ sq, float* __restrict__ bw) {
  __shared__ float s_col[DDIM];
  __shared__ float s_red[256];
  __shared__ float s_tot;
  const int b = blockIdx.x;
  const int tid = threadIdx.x;

  if (tid < DDIM) s_col[tid] = 0.0f;
  __syncthreads();

  float colsum[DDIM];
#pragma unroll
  for (int d = 0; d < DDIM; ++d) colsum[d] = 0.0f;
  float sumsq = 0.0f;

  for (int i = tid; i < NTOT; i += 256) {
    const float* rp = (i < NHALF)
        ? (x + ((size_t)b * NHALF + i) * DDIM)
        : (y + ((size_t)b * NHALF + (i - NHALF)) * DDIM);
    const float4* r4 = (const float4*)rp;
    float s = 0.0f;
#pragma unroll
    for (int q = 0; q < 16; ++q) {
      float4 f = r4[q];
      s += f.x * f.x + f.y * f.y + f.z * f.z + f.w * f.w;
      colsum[q * 4 + 0] += f.x;
      colsum[q * 4 + 1] += f.y;
      colsum[q * 4 + 2] += f.z;
      colsum[q * 4 + 3] += f.w;
    }
    sq[(size_t)b * NTOT + i] = s;
    sumsq += s;
  }

#pragma unroll
  for (int d = 0; d < DDIM; ++d) atomicAdd(&s_col[d], colsum[d]);  // ds_add_f32

  s_red[tid] = sumsq;
  __syncthreads();
  for (int st = 128; st > 0; st >>= 1) {
    if (tid < st) s_red[tid] += s_red[tid + st];
    __syncthreads();
  }
  if (tid == 0) s_tot = s_red[0];
  __syncthreads();

  if (tid < DDIM) s_red[tid] = s_col[tid] * s_col[tid];
  __syncthreads();
  for (int st = 32; st > 0; st >>= 1) {
    if (tid < st) s_red[tid] += s_red[tid + st];
    __syncthreads();
  }
  if (tid == 0) {
    const float N = (float)NTOT;
    const float sumL2 = 2.0f * N * s_tot - 2.0f * s_red[0];
    const float band = sumL2 / (N * N - N);
    bw[b] = band * 0.25f;  // / KERNEL_MUL^(KERNEL_NUM/2) = / 2^2
  }
}

// ---------------------------------------------------------------------------
// Kernel 2: fused Gram(WMMA f32) + 5-bandwidth Gaussian kernel + signed sum.
// 128x64 macro-tile per workgroup; 8 waves (4x2), each wave 32x32 region =
// 2x2 tiles of v_wmma_f32_16x16x4_f32 over K=64 (16 unrolled K-steps).
// ---------------------------------------------------------------------------
__global__ __launch_bounds__(256)
void mmd_gram(const float* __restrict__ x, const float* __restrict__ y,
              const float* __restrict__ sq, const float* __restrict__ bw,
              float* __restrict__ partials) {
  __shared__ float As[TILE_M * LDS_STRIDE];
  __shared__ float Bs[TILE_N * LDS_STRIDE];
  __shared__ float s_wave[8];

  const int b   = blockIdx.z;
  const int tM  = blockIdx.x;  // 0..31 : 128-row slab
  const int tN  = blockIdx.y;  // 0..63 : 64-col slab
  const int tid = threadIdx.x;

  const int rowA0 = tM * TILE_M;
  const int rowB0 = tN * TILE_N;
  const float* baseA = (rowA0 < NHALF)
      ? (x + ((size_t)b * NHALF + rowA0) * DDIM)
      : (y + ((size_t)b * NHALF + (rowA0 - NHALF)) * DDIM);
  const float* baseB = (rowB0 < NHALF)
      ? (x + ((size_t)b * NHALF + rowB0) * DDIM)
      : (y + ((size_t)b * NHALF + (rowB0 - NHALF)) * DDIM);

  // Stage A (128 rows) and B (64 rows) into LDS, float4 granularity.
  {
    const int r = tid >> 1, h = tid & 1;  // 128 rows x 2 halves
    const float4* g = (const float4*)(baseA + (size_t)r * DDIM + h * 32);
    float4* s = (float4*)(&As[r * LDS_STRIDE + h * 32]);
#pragma unroll
    for (int v = 0; v < 8; ++v) s[v] = g[v];
  }
  {
    const int r = tid >> 2, q = tid & 3;  // 64 rows x 4 quarters
    const float4* g = (const float4*)(baseB + (size_t)r * DDIM + q * 16);
    float4* s = (float4*)(&Bs[r * LDS_STRIDE + q * 16]);
#pragma unroll
    for (int v = 0; v < 4; ++v) s[v] = g[v];
  }
  __syncthreads();

  const int w    = tid >> 5;   // wave 0..7
  const int lane = tid & 31;
  const int wm   = w >> 1;     // 0..3 : 32-row sub-slab
  const int wn   = w & 1;      // 0..1 : 32-col sub-slab
  const int lr   = lane & 15;  // row/col within 16x16 tile
  const int kh   = lane >> 4;  // K-half select (A/B 32-bit fragment layout)

  const float* Arow0 = &As[(wm * 32 + lr) * LDS_STRIDE];
  const float* Arow1 = &As[(wm * 32 + 16 + lr) * LDS_STRIDE];
  const float* Brow0 = &Bs[(wn * 32 + lr) * LDS_STRIDE];
  const float* Brow1 = &Bs[(wn * 32 + 16 + lr) * LDS_STRIDE];

  v8f acc[2][2] = {};

#pragma unroll
  for (int ks = 0; ks < 16; ++ks) {
    const int k0 = ks * 4 + kh * 2;  // lane l holds K = 2*(l/16) + {0,1}
    v2f a0 = *(const v2f*)(Arow0 + k0);
    v2f a1 = *(const v2f*)(Arow1 + k0);
    v2f b0 = *(const v2f*)(Brow0 + k0);
    v2f b1 = *(const v2f*)(Brow1 + k0);
    acc[0][0] = __builtin_amdgcn_wmma_f32_16x16x4_f32(false, a0, false, b0,
                                                      (short)0, acc[0][0], false, false);
    acc[0][1] = __builtin_amdgcn_wmma_f32_16x16x4_f32(false, a0, false, b1,
                                                      (short)0, acc[0][1], false, false);
    acc[1][0] = __builtin_amdgcn_wmma_f32_16x16x4_f32(false, a1, false, b0,
                                                      (short)0, acc[1][0], false, false);
    acc[1][1] = __builtin_amdgcn_wmma_f32_16x16x4_f32(false, a1, false, b1,
                                                      (short)0, acc[1][1], false, false);
  }

  // Fused epilogue: L2 -> 5-bandwidth Gaussian kernels -> signed accumulate.
  const float invbw = 1.0f / bw[b];
  const float sgn = ((tM < 16) == (tN < 32)) ? 1.0f : -1.0f;  // s_i * s_j (tile-uniform)
  const float* sqb = sq + (size_t)b * NTOT;
  const int iBase = tM * TILE_M + wm * 32;  // + tm*16 + kh*8 + r
  const int jBase = tN * TILE_N + wn * 32;  // + tn*16 + lr

  float lsum = 0.0f;
#pragma unroll
  for (int tm = 0; tm < 2; ++tm) {
#pragma unroll
    for (int tn = 0; tn < 2; ++tn) {
      const float sqj = sqb[jBase + tn * 16 + lr];
#pragma unroll
      for (int r = 0; r < 8; ++r) {  // C/D layout: M = r + 8*(lane/16), N = lane%16
        const float sqi = sqb[iBase + tm * 16 + kh * 8 + r];
        const float l2 = sqi + sqj - 2.0f * acc[tm][tn][r];
        const float t = -l2 * invbw;
        lsum += __expf(t) + __expf(0.5f * t) + __expf(0.25f * t)
              + __expf(0.125f * t) + __expf(0.0625f * t);
      }
    }
  }
  lsum *= sgn;

  // Deterministic reduction: wave shuffle -> LDS -> one partial per workgroup.
  for (int off = 16; off > 0; off >>= 1) lsum += __shfl_down(lsum, off, 32);
  if (lane == 0) s_wave[w] = lsum;
  __syncthreads();
  if (tid == 0) {
    float tot = 0.0f;
#pragma unroll
    for (int i = 0; i < 8; ++i) tot += s_wave[i];
    partials[((size_t)b * 32 + tM) * 64 + tN] = tot;
  }
}

// ---------------------------------------------------------------------------
// Kernel 3: deterministic final reduction of 2048 partials per batch.
// ---------------------------------------------------------------------------
__global__ __launch_bounds__(256)
void mmd_finalize(const float* __restrict__ partials, float* __restrict__ out) {
  __shared__ float s[256];
  const int b = blockIdx.x;
  const int tid = threadIdx.x;
  float v = 0.0f;
  for (int i = tid; i < 2048; i += 256) v += partials[(size_t)b * 2048 + i];
  s[tid] = v;
  __syncthreads();
  for (int st = 128; st > 0; st >>= 1) {
    if (tid < st) s[tid] += s[tid + st];
    __syncthreads();
  }
  if (tid == 0) out[b] = s[0] * (1.0f / ((float)NHALF * (float)NHALF));
}

extern "C" void kernel_launch(void* const* d_in, const int* in_sizes, int n_in,
                              void* d_out, int out_size, void* d_ws, size_t ws_size,
                              hipStream_t stream) {
  const float* x = (const float*)d_in[0];  // input  (4,2048,64) f32
  const float* y = (const float*)d_in[1];  // target (4,2048,64) f32
  float* out = (float*)d_out;              // (4,) f32
  float* ws = (float*)d_ws;

  float* sq       = ws;              // 4*4096 floats
  float* bw       = ws + 16384;      // 4 floats
  float* partials = ws + 16384 + 4;  // 4*2048 floats

  mmd_prep<<<dim3(4), dim3(256), 0, stream>>>(x, y, sq, bw);
  mmd_gram<<<dim3(32, 64, 4), dim3(256), 0, stream>>>(x, y, sq, bw, partials);
  mmd_finalize<<<dim3(4), dim3(256), 0, stream>>>(partials, out);
}